// Spectrum_24970939859722
// MI455X (gfx1250) — compile-verified
//
#include <hip/hip_runtime.h>
#include <hip/hip_bf16.h>

// Flux-conserving spectral rebin as a banded matmul on the CDNA5 matrix core.
// Per 16-output-bin tile: D(16x16) += A(16x4 f32 flux) x B(4x16 f32 overlap)
// via V_WMMA_F32_16X16X4_F32, sweeping K over the ~130-bin overlap band.
// Fully branchless inner loop: boundary edges are precomputed uniform scalars,
// out-of-band / out-of-array K slices self-mask through zero overlap weights.

typedef __attribute__((ext_vector_type(2))) float v2f;
typedef __attribute__((ext_vector_type(8))) float v8f;

__device__ __forceinline__ int clampi(int i, int lo, int hi) {
    return min(max(i, lo), hi);
}

// edges[i] of monotone grid wl[0..n-1] (i in Z, clamped semantics):
// interior 0.5*(wl[i-1]+wl[i]); i<=0 -> e_first; i>=n -> e_last.
// e_first/e_last passed in as precomputed uniform scalars. All loads clamped.
__device__ __forceinline__ float grid_edge_bl(const float* __restrict__ wl, int i, int n,
                                              float e_first, float e_last) {
    float wa = wl[clampi(i - 1, 0, n - 1)];
    float wb = wl[clampi(i,     0, n - 1)];
    float e = 0.5f * (wa + wb);
    e = (i <= 0) ? e_first : e;
    e = (i >= n) ? e_last  : e;
    return e;
}

__global__ __launch_bounds__(32) void rebin_conserve_wmma(
    const float* __restrict__ wl_in,    // (n_in)
    const float* __restrict__ flux_in,  // (n_spec, n_in)
    const float* __restrict__ wl_out,   // (n_out)
    float* __restrict__ out,            // (n_spec, n_out)
    int n_in, int n_out, int n_spec) {
    const int lane = threadIdx.x;          // one wave per block
    const int o0   = blockIdx.x * 16;      // first output bin of this tile

    // ---- uniform boundary edge values (extrapolated half-steps) ----
    const float in_first  = 1.5f * wl_in[0]          - 0.5f * wl_in[1];
    const float in_last   = 1.5f * wl_in[n_in - 1]   - 0.5f * wl_in[n_in - 2];
    const float out_first = 1.5f * wl_out[0]         - 0.5f * wl_out[1];
    const float out_last  = 1.5f * wl_out[n_out - 1] - 0.5f * wl_out[n_out - 2];

    // ---- per-lane output-bin interval (lane l owns B/C/D column n = l & 15) ----
    const int  n       = lane & 15;
    const int  o       = o0 + n;
    const bool o_valid = (o < n_out);
    // invalid columns collapse to a zero-width interval -> overlap weights = 0
    const float eo_lo = o_valid ? grid_edge_bl(wl_out, o,     n_out, out_first, out_last) : out_last;
    const float eo_hi = o_valid ? grid_edge_bl(wl_out, o + 1, n_out, out_first, out_last) : out_last;
    const float rdelta_out = o_valid ? 1.0f / (eo_hi - eo_lo) : 0.0f;

    // ---- band extent for this tile (uniform across the wave) ----
    const float tile_lo = grid_edge_bl(wl_out, o0, n_out, out_first, out_last);
    const float tile_hi = grid_edge_bl(wl_out, min(o0 + 16, n_out), n_out, out_first, out_last);

    // k0: first input bin whose high edge exceeds tile_lo (uniform binary search)
    int lo = 0, hi = n_in - 1;
    while (lo < hi) {
        int mid = (lo + hi) >> 1;
        if (grid_edge_bl(wl_in, mid + 1, n_in, in_first, in_last) > tile_lo) hi = mid;
        else lo = mid + 1;
    }
    const int k0 = lo;
    // k_end: first input bin whose low edge reaches tile_hi (uniform binary search)
    lo = k0; hi = n_in;
    while (lo < hi) {
        int mid = (lo + hi) >> 1;
        if (grid_edge_bl(wl_in, mid, n_in, in_first, in_last) >= tile_hi) hi = mid;
        else lo = mid + 1;
    }
    const int k_end = lo;

    // ---- fragment index roles for this lane ----
    // A (16x4 f32, MxK): lanes 0-15 -> M=lane, K={0,1}; lanes 16-31 -> M=lane-16, K={2,3}
    // B (4x16 f32, KxN): lanes 0-15 -> N=lane, K={0,1}; lanes 16-31 -> N=lane-16, K={2,3}
    const int   m        = lane & 15;
    const int   kof      = (lane < 16) ? 0 : 2;
    const float row_mask = (m < n_spec) ? 1.0f : 0.0f;           // loop-invariant row mask
    const int   flux_row = clampi(m, 0, n_spec - 1) * n_in;      // clamped, always in-bounds

    // builds one K-chunk-of-4 fragment pair and accumulates with one WMMA
    auto chunk = [&](int k, v8f c) -> v8f {
        const int i0 = k + kof;               // this lane's first K slice
        // three consecutive input-bin edges (clamped loads; boundary via cndmask)
        const float e0 = grid_edge_bl(wl_in, i0,     n_in, in_first, in_last);
        const float e1 = grid_edge_bl(wl_in, i0 + 1, n_in, in_first, in_last);
        const float e2 = grid_edge_bl(wl_in, i0 + 2, n_in, in_first, in_last);
        // B fragment: overlap lengths (self-masking outside band/array)
        v2f b;
        b.x = fmaxf(fminf(eo_hi, e1) - fmaxf(eo_lo, e0), 0.0f);
        b.y = fmaxf(fminf(eo_hi, e2) - fmaxf(eo_lo, e1), 0.0f);
        // A fragment: flux (clamped address; validity enforced by B's zeros)
        v2f a;
        a.x = row_mask * flux_in[flux_row + clampi(i0,     0, n_in - 1)];
        a.y = row_mask * flux_in[flux_row + clampi(i0 + 1, 0, n_in - 1)];
        return __builtin_amdgcn_wmma_f32_16x16x4_f32(
            /*neg_a=*/false, a, /*neg_b=*/false, b,
            /*c_mod=*/(short)0, c, /*reuse_a=*/false, /*reuse_b=*/false);
    };

    // ---- sweep the band, 8 K slices (2 WMMAs) per iteration, 2 accumulators ----
    v8f c0 = {}, c1 = {};
    for (int k = k0; k < k_end; k += 8) {       // overshoot past k_end is self-masked
        c0 = chunk(k,     c0);
        c1 = chunk(k + 4, c1);
    }
    v8f c = c0 + c1;   // D holds flux_out * delta_out

    // ---- store: C/D layout — VGPR v holds M=v (lanes 0-15) or M=v+8 (lanes 16-31) ----
    if (o_valid) {
        const int m_base = (lane < 16) ? 0 : 8;
        #pragma unroll
        for (int v = 0; v < 8; ++v) {
            const int s = m_base + v;
            if (s < n_spec) {
                out[s * n_out + o] = c[v] * rdelta_out;
            }
        }
    }
}

extern "C" void kernel_launch(void* const* d_in, const int* in_sizes, int n_in_arrs,
                              void* d_out, int out_size, void* d_ws, size_t ws_size,
                              hipStream_t stream) {
    (void)n_in_arrs; (void)out_size; (void)d_ws; (void)ws_size;
    const float* wl_in   = (const float*)d_in[0];
    const float* flux_in = (const float*)d_in[1];
    const float* wl_out  = (const float*)d_in[2];
    float* out           = (float*)d_out;

    const int n_in   = in_sizes[0];
    const int n_out  = in_sizes[2];
    const int n_spec = in_sizes[1] / n_in;

    const int n_tiles = (n_out + 15) / 16;
    rebin_conserve_wmma<<<dim3(n_tiles), dim3(32), 0, stream>>>(
        wl_in, flux_in, wl_out, out, n_in, n_out, n_spec);
}